// Ensemble_attention_24472723653290
// MI455X (gfx1250) — compile-verified
//
#include <hip/hip_runtime.h>
#include <hip/hip_bf16.h>

#define BDIM 64
#define PDIM 196
#define DDIM 768
#define XSTRIDE (197 * 768)   // per-b stride of x (B, P+1, DIM)

typedef __attribute__((ext_vector_type(2))) float v2f;
typedef __attribute__((ext_vector_type(8))) float v8f;

__device__ __forceinline__ int swz(int row, int col) {
    // XOR swizzle: keeps float4 / b64 alignment (mask is a multiple of 4),
    // makes A-operand ds_load_b64 bank-conflict-free (768 % 64 == 0 otherwise).
    return row * DDIM + (col ^ ((row & 15) << 2));
}

// One workgroup per p: S = G(64x768) @ W_p(768x768) via V_WMMA_F32_16X16X4_F32,
// fused with the row-wise dot against embeds_local to produce attn logits.
__global__ __launch_bounds__(256)
void attn_logits_kernel(const float* __restrict__ x,
                        const float* __restrict__ W,
                        float* __restrict__ logits) {
    __shared__ float gshared[BDIM * DDIM];   // 192 KB (fits CDNA5 320KB LDS/WGP)
    __shared__ float attn_acc[BDIM];

    const int p   = blockIdx.x;
    const int tid = threadIdx.x;

    // ---- stage G = x[:,0,:] into LDS (swizzled), float4 granularity ----
    for (int i = tid; i < BDIM * DDIM / 4; i += 256) {
        const int li  = i * 4;
        const int row = li / DDIM;
        const int col = li % DDIM;
        const float4 v = *reinterpret_cast<const float4*>(&x[(size_t)row * XSTRIDE + col]);
        *reinterpret_cast<float4*>(&gshared[swz(row, col)]) = v;
    }
    if (tid < BDIM) attn_acc[tid] = 0.0f;
    __syncthreads();

    const int wave = tid >> 5;
    const int lane = tid & 31;
    const int sub  = lane >> 4;   // half-wave select
    const int idx  = lane & 15;

    const float* Wp = W + (size_t)p * DDIM * DDIM;
    const float* Lp = x + (size_t)(p + 1) * DDIM;   // embeds_local[b][p][e] = Lp[b*XSTRIDE + e]

    for (int j = 0; j < 6; ++j) {
        const int n0 = (wave * 6 + j) * 16;
        v8f acc0 = {}, acc1 = {}, acc2 = {}, acc3 = {};
        const float* wcol = Wp + n0 + idx;

        for (int k = 0; k < DDIM; k += 4) {
            const int k2 = k + 2 * sub;           // VGPR0 <-> K even (lanes16-31: K+2)
            v2f b;
            b.x = wcol[(size_t)k2 * DDIM];
            b.y = wcol[(size_t)(k2 + 1) * DDIM];

            v2f a0 = *reinterpret_cast<const v2f*>(&gshared[swz(0 * 16 + idx, k2)]);
            acc0 = __builtin_amdgcn_wmma_f32_16x16x4_f32(false, a0, false, b, (short)0, acc0, false, false);
            v2f a1 = *reinterpret_cast<const v2f*>(&gshared[swz(1 * 16 + idx, k2)]);
            acc1 = __builtin_amdgcn_wmma_f32_16x16x4_f32(false, a1, false, b, (short)0, acc1, false, false);
            v2f a2 = *reinterpret_cast<const v2f*>(&gshared[swz(2 * 16 + idx, k2)]);
            acc2 = __builtin_amdgcn_wmma_f32_16x16x4_f32(false, a2, false, b, (short)0, acc2, false, false);
            v2f a3 = *reinterpret_cast<const v2f*>(&gshared[swz(3 * 16 + idx, k2)]);
            acc3 = __builtin_amdgcn_wmma_f32_16x16x4_f32(false, a3, false, b, (short)0, acc3, false, false);
        }

        // fused dot with embeds_local: C layout -> lane: N = n0+idx, M = r + 8*sub
        const float* lcol = Lp + n0 + idx;
        #pragma unroll
        for (int r = 0; r < 8; ++r) {
            #pragma unroll
            for (int m = 0; m < 4; ++m) {
                const int bRow = m * 16 + r + 8 * sub;
                float cv = (m == 0) ? acc0[r] : (m == 1) ? acc1[r] : (m == 2) ? acc2[r] : acc3[r];
                float v = cv * lcol[(size_t)bRow * XSTRIDE];
                // reduce across the 16 lanes of this half-wave (same bRow)
                v += __shfl_xor(v, 1, 32);
                v += __shfl_xor(v, 2, 32);
                v += __shfl_xor(v, 4, 32);
                v += __shfl_xor(v, 8, 32);
                if (idx == 0) atomicAdd(&attn_acc[bRow], v);   // ds_add_f32
            }
        }
    }
    __syncthreads();

    const float SCALE = 0.03608439182435161f;   // 768^-0.5
    if (tid < BDIM) logits[(size_t)tid * PDIM + p] = attn_acc[tid] * SCALE;
}

// One block per b: softmax over P=196 logits, then embed_global + sum_p w_p * L[b,p,:]
__global__ __launch_bounds__(256)
void softmax_combine_kernel(const float* __restrict__ x,
                            const float* __restrict__ logits,
                            float* __restrict__ out) {
    __shared__ float wts[PDIM];
    __shared__ float red[8];
    const int b   = blockIdx.x;
    const int tid = threadIdx.x;

    float v = (tid < PDIM) ? logits[(size_t)b * PDIM + tid] : -3.4e38f;

    float m = v;
    m = fmaxf(m, __shfl_xor(m, 16, 32));
    m = fmaxf(m, __shfl_xor(m, 8, 32));
    m = fmaxf(m, __shfl_xor(m, 4, 32));
    m = fmaxf(m, __shfl_xor(m, 2, 32));
    m = fmaxf(m, __shfl_xor(m, 1, 32));
    if ((tid & 31) == 0) red[tid >> 5] = m;
    __syncthreads();
    float bm = red[0];
    #pragma unroll
    for (int i = 1; i < 8; ++i) bm = fmaxf(bm, red[i]);

    float e = (tid < PDIM) ? __expf(v - bm) : 0.0f;
    float s = e;
    s += __shfl_xor(s, 16, 32);
    s += __shfl_xor(s, 8, 32);
    s += __shfl_xor(s, 4, 32);
    s += __shfl_xor(s, 2, 32);
    s += __shfl_xor(s, 1, 32);
    __syncthreads();
    if ((tid & 31) == 0) red[tid >> 5] = s;
    __syncthreads();
    float bs = 0.0f;
    #pragma unroll
    for (int i = 0; i < 8; ++i) bs += red[i];

    if (tid < PDIM) wts[tid] = e / bs;
    __syncthreads();

    const float* xb = x + (size_t)b * XSTRIDE;
    for (int d = tid; d < DDIM; d += 256) {
        float acc = xb[d];                      // embed_global[b][d]
        const float* lrow = xb + DDIM + d;      // x[b][1+p][d]
        #pragma unroll 4
        for (int pp = 0; pp < PDIM; ++pp)
            acc += wts[pp] * lrow[(size_t)pp * DDIM];
        out[(size_t)b * DDIM + d] = acc;
    }
}

extern "C" void kernel_launch(void* const* d_in, const int* in_sizes, int n_in,
                              void* d_out, int out_size, void* d_ws, size_t ws_size,
                              hipStream_t stream) {
    const float* x = (const float*)d_in[0];        // (64, 197, 768) f32
    const float* W = (const float*)d_in[1];        // (196, 768, 768) f32
    float* out     = (float*)d_out;                // (64, 768) f32
    float* logits  = (float*)d_ws;                 // 64*196 f32 scratch

    attn_logits_kernel<<<PDIM, 256, 0, stream>>>(x, W, logits);
    softmax_combine_kernel<<<BDIM, 256, 0, stream>>>(x, logits, out);
}